// RPSNet_weights_6219112645218
// MI455X (gfx1250) — compile-verified
//
#include <hip/hip_runtime.h>
#include <math.h>

typedef float v2f __attribute__((ext_vector_type(2)));
typedef float v8f __attribute__((ext_vector_type(8)));

// ---------------------------------------------------------------------------
// Fused kernel: y = x @ W^T via V_WMMA_F32_16X16X4_F32, then per-lane
// symmetric-reduced damped Newton (4x4 solve) for the logit QRE.
// One wave = 32 batch rows (two 16-row WMMA tiles); lane L owns row base+L.
// ---------------------------------------------------------------------------
__global__ __launch_bounds__(256) void qre_fused_kernel(
    const float* __restrict__ x, const float* __restrict__ W,
    float* __restrict__ out_u, float* __restrict__ out_v,
    float* __restrict__ out_temp, int F)
{
    __shared__ float ylds[8][32][3];

    const int lane = threadIdx.x & 31;
    const int wave = threadIdx.x >> 5;
    const int m    = lane & 15;      // M (A tiles) / N (B tile)
    const int hi   = lane >> 4;      // which K-half of the k0..k0+3 chunk
    const long rowBase = (long)blockIdx.x * 256 + (long)wave * 32;

    // A fragments: lane holds x[row][k0+2*hi], x[row][k0+2*hi+1]  (float2)
    const float* a0p = x + (rowBase + m) * (long)F + 2 * hi;
    const float* a1p = x + (rowBase + 16 + m) * (long)F + 2 * hi;
    // B fragment: lane N holds W[N][k0+2*hi .. +1]; N>=3 -> zero (branchless)
    const float  bsc = (m < 3) ? 1.0f : 0.0f;
    const float* bp  = W + (long)((m < 3) ? m : 2) * F + 2 * hi;

    v8f c0 = {}; v8f c1 = {};
    #pragma unroll 4
    for (int k0 = 0; k0 < F; k0 += 4) {
        v2f a0 = *(const v2f*)(a0p + k0);
        v2f a1 = *(const v2f*)(a1p + k0);
        v2f b  = *(const v2f*)(bp + k0);
        b.x *= bsc; b.y *= bsc;
        c0 = __builtin_amdgcn_wmma_f32_16x16x4_f32(false, a0, false, b,
                                                   (short)0, c0, false, false);
        c1 = __builtin_amdgcn_wmma_f32_16x16x4_f32(false, a1, false, b,
                                                   (short)0, c1, false, false);
    }

    // C layout: lane holds column N=m for rows M = vgpr + 8*hi.
    if (m < 3) {
        #pragma unroll
        for (int r = 0; r < 8; ++r) {
            const int M = r + 8 * hi;
            ylds[wave][M][m]      = c0[r];
            ylds[wave][16 + M][m] = c1[r];
        }
    }
    __syncthreads();

    const float y0 = ylds[wave][lane][0];
    const float y1 = ylds[wave][lane][1];
    const float y2 = ylds[wave][lane][2];

    // P = [[0, y0, -y1], [-y0, 0, y2], [y1, -y2, 0]]  (SCALE = 1)
    // Symmetric QRE: u == v for skew-symmetric P; reduced KKT system:
    //   [diag(1/u)+P  1] [du ]   [-(P u + log u + 1 + mu)]
    //   [    1^T      0] [dmu] = [-(sum(u) - 1)          ]
    float u0 = 1.0f / 3.0f, u1 = 1.0f / 3.0f, u2 = 1.0f / 3.0f, mu = 0.0f;

    #pragma unroll 1
    for (int it = 0; it < 30; ++it) {
        const float F0 = y0 * u1 - y1 * u2 + logf(u0) + 1.0f + mu;
        const float F1 = -y0 * u0 + y2 * u2 + logf(u1) + 1.0f + mu;
        const float F2 = y1 * u0 - y2 * u1 + logf(u2) + 1.0f + mu;
        const float g  = u0 + u1 + u2 - 1.0f;

        float M4[4][5] = {
            { 1.0f / u0,  y0,        -y1,        1.0f, -F0 },
            { -y0,        1.0f / u1,  y2,        1.0f, -F1 },
            { y1,        -y2,         1.0f / u2, 1.0f, -F2 },
            { 1.0f,       1.0f,       1.0f,      0.0f, -g  }
        };

        // Fully unrolled partial-pivot Gaussian elimination (stays in regs).
        #pragma unroll
        for (int k = 0; k < 4; ++k) {
            #pragma unroll
            for (int p = k + 1; p < 4; ++p) {
                const bool sw = fabsf(M4[p][k]) > fabsf(M4[k][k]);
                #pragma unroll
                for (int j = k; j < 5; ++j) {
                    const float a = M4[k][j], b = M4[p][j];
                    M4[k][j] = sw ? b : a;
                    M4[p][j] = sw ? a : b;
                }
            }
            const float inv = 1.0f / M4[k][k];
            #pragma unroll
            for (int i = k + 1; i < 4; ++i) {
                const float f = M4[i][k] * inv;
                #pragma unroll
                for (int j = k + 1; j < 5; ++j) M4[i][j] -= f * M4[k][j];
            }
        }
        const float d3 = M4[3][4] / M4[3][3];
        const float d2 = (M4[2][4] - M4[2][3] * d3) / M4[2][2];
        const float d1 = (M4[1][4] - M4[1][2] * d2 - M4[1][3] * d3) / M4[1][1];
        const float d0 = (M4[0][4] - M4[0][1] * d1 - M4[0][2] * d2 - M4[0][3] * d3) / M4[0][0];

        // Positivity-preserving damped step (ratios duplicate across u==v).
        float tl = __builtin_inff();
        tl = (d0 < 0.0f) ? fminf(tl, -u0 / d0) : tl;
        tl = (d1 < 0.0f) ? fminf(tl, -u1 / d1) : tl;
        tl = (d2 < 0.0f) ? fminf(tl, -u2 / d2) : tl;
        const float t = fminf(1.0f, 0.9f * tl);

        u0 += t * d0; u1 += t * d1; u2 += t * d2; mu += t * d3;
    }

    const long r = rowBase + lane;
    out_u[r * 3 + 0] = u0; out_u[r * 3 + 1] = u1; out_u[r * 3 + 2] = u2;
    out_v[r * 3 + 0] = u0; out_v[r * 3 + 1] = u1; out_v[r * 3 + 2] = u2;

    float* tp = out_temp + r * 9;
    tp[0] = 0.0f; tp[1] = y0;   tp[2] = -y1;
    tp[3] = -y0;  tp[4] = 0.0f; tp[5] = y2;
    tp[6] = y1;   tp[7] = -y2;  tp[8] = 0.0f;
}

// ---------------------------------------------------------------------------
// W_exp broadcast: one block per batch row, 384 threads x float4 (coalesced).
// This is the HBM-store-bound part (~402 MB).
// ---------------------------------------------------------------------------
__global__ void wexp_kernel(const float4* __restrict__ W4,
                            float4* __restrict__ out4, int row4)
{
    const long idx = (long)blockIdx.x * row4 + threadIdx.x;
    out4[idx] = W4[threadIdx.x];
}

extern "C" void kernel_launch(void* const* d_in, const int* in_sizes, int n_in,
                              void* d_out, int out_size, void* d_ws, size_t ws_size,
                              hipStream_t stream)
{
    const float* x = (const float*)d_in[0];
    const float* W = (const float*)d_in[1];
    const int  F = in_sizes[1] / 3;          // 512
    const long B = (long)in_sizes[0] / F;    // 65536

    float* out      = (float*)d_out;
    float* out_u    = out;                   // (B,3)
    float* out_v    = out + 3 * B;           // (B,3)
    float* out_temp = out + 6 * B;           // (B,3,3)
    float* out_W    = out + 15 * B;          // (B,3,F)

    qre_fused_kernel<<<(unsigned)(B / 256), 256, 0, stream>>>(
        x, W, out_u, out_v, out_temp, F);

    const int row4 = (3 * F) / 4;            // 384 float4 per batch row
    wexp_kernel<<<(unsigned)B, row4, 0, stream>>>(
        (const float4*)W, (float4*)out_W, row4);
}